// GCN_3607772528647
// MI455X (gfx1250) — compile-verified
//
#include <hip/hip_runtime.h>

typedef __attribute__((ext_vector_type(2))) float v2f;
typedef __attribute__((ext_vector_type(4))) float v4f;
typedef __attribute__((ext_vector_type(8))) float v8f;

#define N_ATOMS 100000
#define N_BONDS 200000
#define MAX_NEI 6
#define N_MOLS  5000
#define HID     256
#define FB      88
#define FA      82
#define BPS     272  // pair-row stride in float2 units: 272*8B = 2176B = +32 banks

__device__ __forceinline__ v8f wmma_f32(v2f a, v2f b, v8f c) {
  // 8 args: (neg_a, A, neg_b, B, c_mod, C, reuse_a, reuse_b)
  return __builtin_amdgcn_wmma_f32_16x16x4_f32(false, a, false, b, (short)0, c,
                                               false, false);
}

// Stage `pairs` K-pairs of a row-major [*, 256] weight slab into LDS with
// pair-interleaved layout: Bp[p*BPS + c] = {W[2p][c], W[2p+1][c]}.
// Rows >= nrows are zero-filled (tail guard).
__device__ __forceinline__ void stage_pairs(v2f* Bp, const float* __restrict__ W,
                                            int nrows, int pairs, int tid) {
  for (int i = tid; i < pairs * 64; i += 256) {
    const int p = i >> 6, c4 = (i & 63) * 4;
    const int r0 = 2 * p;
    v4f lo = {}, hi = {};
    if (r0 < nrows)     lo = *(const v4f*)(W + (size_t)r0 * HID + c4);
    if (r0 + 1 < nrows) hi = *(const v4f*)(W + (size_t)(r0 + 1) * HID + c4);
    v4f e0 = {lo.x, hi.x, lo.y, hi.y};
    v4f e1 = {lo.z, hi.z, lo.w, hi.w};
    *(v4f*)(Bp + (size_t)p * BPS + c4) = e0;
    *(v4f*)(Bp + (size_t)p * BPS + c4 + 2) = e1;
  }
}

// One K-chunk of WMMA work. Wave tile = 32 rows x 64 cols:
// 2 A-frag ds_load_b64 + 4 B-frag ds_load_b64 -> 8 v_wmma per k-step.
__device__ __forceinline__ void gemm_chunk(const v2f* __restrict__ Bp,
                                           const float* __restrict__ A0,
                                           const float* __restrict__ A1,
                                           int steps, int nbl, int half,
                                           v8f acc[2][4]) {
  for (int s = 0; s < steps; ++s) {
    const int k = s * 4;
    v2f a0 = *(const v2f*)(A0 + k);
    v2f a1 = *(const v2f*)(A1 + k);
    const v2f* brow = Bp + (size_t)(2 * s + half) * BPS + nbl;
#pragma unroll
    for (int t = 0; t < 4; ++t) {
      v2f b = brow[t * 16];
      acc[0][t] = wmma_f32(a0, b, acc[0][t]);
      acc[1][t] = wmma_f32(a1, b, acc[1][t]);
    }
  }
}

// ---------------------------------------------------------------------------
// K1: nei_input_h = fbonds @ W_i ; msg = relu(nei_input_h)
// block = 64 rows x 256 cols, 8 waves, wave = 32 rows x 64 cols.
// W_i (88x256) staged once into LDS as 44 interleaved pairs.
// ---------------------------------------------------------------------------
__global__ __launch_bounds__(256) void k_bond_in(
    const float* __restrict__ fbonds, const float* __restrict__ Wi,
    float* __restrict__ nei_input, float* __restrict__ msg) {
  __shared__ v2f Bp[44 * BPS];  // 95.7 KB
  const int tid = threadIdx.x;
  const int wave = tid >> 5, lane = tid & 31;
  const int half = lane >> 4, l16 = lane & 15;
  const int rs = (wave >> 2) * 32;
  const int nb = (wave & 3) * 64;
  const int m0 = blockIdx.x * 64;

  stage_pairs(Bp, Wi, FB, 44, tid);
  __syncthreads();

  v8f acc[2][4] = {};
  const float* A0 = fbonds + (size_t)(m0 + rs + l16) * FB + half * 2;
  const float* A1 = A0 + (size_t)16 * FB;
  gemm_chunk(Bp, A0, A1, FB / 4, nb + l16, half, acc);

#pragma unroll
  for (int m = 0; m < 2; ++m)
#pragma unroll
    for (int t = 0; t < 4; ++t) {
      const int col = nb + t * 16 + l16;
#pragma unroll
      for (int r = 0; r < 8; ++r) {
        const size_t idx = (size_t)(m0 + rs + m * 16 + r + half * 8) * HID + col;
        float v = acc[m][t][r];
        nei_input[idx] = v;
        msg[idx] = fmaxf(v, 0.f);
      }
    }
}

// ---------------------------------------------------------------------------
// K2: msg_out = relu(nei_input + (sum_j msg_in[bgraph[:,j]]) @ W_h)
// Fused gather -> LDS A (64x260) ; W_h streamed in 64-row LDS chunks.
// ---------------------------------------------------------------------------
#define LDA2 260
__global__ __launch_bounds__(256) void k_layer(
    const float* __restrict__ msg_in, const int* __restrict__ bgraph,
    const float* __restrict__ Wh, const float* __restrict__ nei_input,
    float* __restrict__ msg_out) {
  __shared__ float As[64 * LDA2];     // 65.0 KB
  __shared__ v2f Bp[32 * BPS];        // 68.0 KB
  __shared__ int nidx[64 * MAX_NEI];  // 1.5 KB
  const int tid = threadIdx.x;
  const int m0 = blockIdx.x * 64;

  for (int i = tid; i < 64 * MAX_NEI; i += 256)
    nidx[i] = bgraph[(size_t)m0 * MAX_NEI + i];
  __syncthreads();

  for (int c = tid; c < 64 * 64; c += 256) {
    const int rr = c >> 6;
    const int c4 = (c & 63) * 4;
    v4f s = {};
#pragma unroll
    for (int j = 0; j < MAX_NEI; ++j)
      s += *(const v4f*)(msg_in + (size_t)nidx[rr * MAX_NEI + j] * HID + c4);
    *(v4f*)(&As[rr * LDA2 + c4]) = s;
  }
  __syncthreads();

  const int wave = tid >> 5, lane = tid & 31;
  const int half = lane >> 4, l16 = lane & 15;
  const int rs = (wave >> 2) * 32;
  const int nb = (wave & 3) * 64;
  v8f acc[2][4] = {};
  const float* A0 = As + (size_t)(rs + l16) * LDA2 + half * 2;
  const float* A1 = A0 + (size_t)16 * LDA2;

  for (int kb = 0; kb < HID; kb += 64) {
    stage_pairs(Bp, Wh + (size_t)kb * HID, HID, 32, tid);
    __syncthreads();
    gemm_chunk(Bp, A0 + kb, A1 + kb, 16, nb + l16, half, acc);
    __syncthreads();
  }

#pragma unroll
  for (int m = 0; m < 2; ++m)
#pragma unroll
    for (int t = 0; t < 4; ++t) {
      const int col = nb + t * 16 + l16;
#pragma unroll
      for (int r = 0; r < 8; ++r) {
        const size_t idx = (size_t)(m0 + rs + m * 16 + r + half * 8) * HID + col;
        msg_out[idx] = fmaxf(acc[m][t][r] + nei_input[idx], 0.f);
      }
    }
}

// ---------------------------------------------------------------------------
// K3: atom_h = concat(fatoms, sum_j msg[agraph[:,j]]) @ W_o + b_o
// LDS A row layout: [nei(256) | fatoms(82) | pad2], stride 340.
// K streamed in 6 chunks: 4x Wo rows 82..337 (vs nei), then Wo rows 0..63
// and rows 64..81(+2 zero) (vs fatoms+pad).
// ---------------------------------------------------------------------------
#define LDA3 340
__global__ __launch_bounds__(256) void k_atom(
    const float* __restrict__ fatoms, const int* __restrict__ agraph,
    const float* __restrict__ msg, const float* __restrict__ Wo,
    const float* __restrict__ bo, float* __restrict__ atom_h) {
  __shared__ float As[64 * LDA3];     // 85.0 KB
  __shared__ v2f Bp[32 * BPS];        // 68.0 KB
  __shared__ int nidx[64 * MAX_NEI];  // 1.5 KB
  const int tid = threadIdx.x;
  const int m0 = blockIdx.x * 64;

  for (int i = tid; i < 64 * MAX_NEI; i += 256) {
    const int row = m0 + i / MAX_NEI;
    nidx[i] = (row < N_ATOMS) ? agraph[(size_t)m0 * MAX_NEI + i] : 0;
  }
  __syncthreads();

  for (int c = tid; c < 64 * 64; c += 256) {
    const int rr = c >> 6;
    const int c4 = (c & 63) * 4;
    v4f s = {};
#pragma unroll
    for (int j = 0; j < MAX_NEI; ++j)
      s += *(const v4f*)(msg + (size_t)nidx[rr * MAX_NEI + j] * HID + c4);
    *(v4f*)(&As[rr * LDA3 + c4]) = s;
  }
  for (int i = tid; i < 64 * FA; i += 256) {
    const int rr = i / FA, cc = i % FA;
    const int row = m0 + rr;
    As[rr * LDA3 + HID + cc] = (row < N_ATOMS) ? fatoms[(size_t)row * FA + cc] : 0.f;
  }
  if (tid < 128) As[(tid >> 1) * LDA3 + HID + FA + (tid & 1)] = 0.f;
  __syncthreads();

  const int wave = tid >> 5, lane = tid & 31;
  const int half = lane >> 4, l16 = lane & 15;
  const int rs = (wave >> 2) * 32;
  const int nb = (wave & 3) * 64;
  v8f acc[2][4] = {};
  const float* A0 = As + (size_t)(rs + l16) * LDA3 + half * 2;
  const float* A1 = A0 + (size_t)16 * LDA3;

  // chunks: {A-col base, Wo row base, pairs, steps, valid rows}
  // seg1: nei cols 0..255 vs Wo rows 82..337
  for (int c = 0; c < 4; ++c) {
    stage_pairs(Bp, Wo + (size_t)(FA + 64 * c) * HID, 64, 32, tid);
    __syncthreads();
    gemm_chunk(Bp, A0 + 64 * c, A1 + 64 * c, 16, nb + l16, half, acc);
    __syncthreads();
  }
  // seg2a: fatoms cols 256..319 vs Wo rows 0..63
  stage_pairs(Bp, Wo, 64, 32, tid);
  __syncthreads();
  gemm_chunk(Bp, A0 + HID, A1 + HID, 16, nb + l16, half, acc);
  __syncthreads();
  // seg2b: cols 320..339 (incl. 2 zero pad) vs Wo rows 64..81 (+2 zeros)
  stage_pairs(Bp, Wo + (size_t)64 * HID, FA - 64, 10, tid);
  __syncthreads();
  gemm_chunk(Bp, A0 + HID + 64, A1 + HID + 64, 5, nb + l16, half, acc);

#pragma unroll
  for (int m = 0; m < 2; ++m)
#pragma unroll
    for (int t = 0; t < 4; ++t) {
      const int col = nb + t * 16 + l16;
      const float bias = bo[col];
#pragma unroll
      for (int r = 0; r < 8; ++r) {
        const int row = m0 + rs + m * 16 + r + half * 8;
        if (row < N_ATOMS)
          atom_h[(size_t)row * HID + col] = acc[m][t][r] + bias;
      }
    }
}

// ---------------------------------------------------------------------------
// K4: mol_in[m] = sum of atom_h rows with mol_ids==m (sorted ids ->
// binary-searched contiguous range; deterministic, no float atomics)
// ---------------------------------------------------------------------------
__global__ __launch_bounds__(64) void k_segsum(
    const float* __restrict__ atom_h, const int* __restrict__ mol_ids,
    float* __restrict__ mol_in) {
  const int mol = blockIdx.x;
  const int tid = threadIdx.x;
  int lo = 0, hi = N_ATOMS;
  while (lo < hi) { int mid = (lo + hi) >> 1; if (mol_ids[mid] < mol) lo = mid + 1; else hi = mid; }
  const int s = lo;
  hi = N_ATOMS;
  while (lo < hi) { int mid = (lo + hi) >> 1; if (mol_ids[mid] < mol + 1) lo = mid + 1; else hi = mid; }
  const int e = lo;
  v4f acc = {};
  for (int a = s; a < e; ++a)
    acc += *(const v4f*)(atom_h + (size_t)a * HID + tid * 4);
  *(v4f*)(mol_in + (size_t)mol * HID + tid * 4) = acc;
}

// ---------------------------------------------------------------------------
// K5: mol_o = relu(mol_in @ W_mh + b_mh) @ W_mo + b_mo  (tiny: 0.7 GFLOP)
// ---------------------------------------------------------------------------
__global__ __launch_bounds__(64) void k_mol(
    const float* __restrict__ mol_in, const float* __restrict__ Wmh,
    const float* __restrict__ bmh, const float* __restrict__ Wmo,
    const float* __restrict__ bmo, float* __restrict__ mol_o) {
  __shared__ float xin[HID];
  __shared__ float red[64];
  const int mol = blockIdx.x, tid = threadIdx.x;
  *(v4f*)(&xin[tid * 4]) = *(const v4f*)(mol_in + (size_t)mol * HID + tid * 4);
  __syncthreads();
  float part = 0.f;
#pragma unroll
  for (int i = 0; i < 4; ++i) {
    const int c = tid * 4 + i;
    float acc = bmh[c];
    for (int k = 0; k < HID; ++k)
      acc = fmaf(xin[k], Wmh[(size_t)k * HID + c], acc);
    part = fmaf(fmaxf(acc, 0.f), Wmo[c], part);
  }
  red[tid] = part;
  __syncthreads();
  for (int sft = 32; sft > 0; sft >>= 1) {
    if (tid < sft) red[tid] += red[tid + sft];
    __syncthreads();
  }
  if (tid == 0) mol_o[mol] = red[0] + bmo[0];
}

// ---------------------------------------------------------------------------
extern "C" void kernel_launch(void* const* d_in, const int* in_sizes, int n_in,
                              void* d_out, int out_size, void* d_ws,
                              size_t ws_size, hipStream_t stream) {
  const float* fatoms  = (const float*)d_in[0];
  const float* fbonds  = (const float*)d_in[1];
  const int*   agraph  = (const int*)d_in[2];
  const int*   bgraph  = (const int*)d_in[3];
  const int*   mol_ids = (const int*)d_in[4];
  const float* W_i     = (const float*)d_in[5];
  const float* W_h     = (const float*)d_in[6];
  const float* W_o     = (const float*)d_in[7];
  const float* b_o     = (const float*)d_in[8];
  const float* W_mh    = (const float*)d_in[9];
  const float* b_mh    = (const float*)d_in[10];
  const float* W_mo    = (const float*)d_in[11];
  const float* b_mo    = (const float*)d_in[12];

  float* ws = (float*)d_ws;
  float* nei_input = ws;                                 // 51.2M floats
  float* msgA = nei_input + (size_t)N_BONDS * HID;       // 51.2M
  float* msgB = msgA + (size_t)N_BONDS * HID;            // 51.2M
  float* mol_in = msgB + (size_t)N_BONDS * HID;          // 1.28M

  float* atom_h = (float*)d_out;                         // [N_ATOMS, HID]
  float* mol_o = (float*)d_out + (size_t)N_ATOMS * HID;  // [N_MOLS]

  k_bond_in<<<N_BONDS / 64, 256, 0, stream>>>(fbonds, W_i, nei_input, msgA);

  float* cur = msgA;
  float* nxt = msgB;
  for (int it = 0; it < 3; ++it) {  // N_LAYERS - 1
    k_layer<<<N_BONDS / 64, 256, 0, stream>>>(cur, bgraph, W_h, nei_input, nxt);
    float* t = cur; cur = nxt; nxt = t;
  }

  k_atom<<<(N_ATOMS + 63) / 64, 256, 0, stream>>>(fatoms, agraph, cur, W_o,
                                                  b_o, atom_h);
  k_segsum<<<N_MOLS, 64, 0, stream>>>(atom_h, mol_ids, mol_in);
  k_mol<<<N_MOLS, 64, 0, stream>>>(mol_in, W_mh, b_mh, W_mo, b_mo, mol_o);
}